// SoftMaxTreePrediction_64570538328434
// MI455X (gfx1250) — compile-verified
//
#include <hip/hip_runtime.h>
#include <cstdint>

// Tree geometry (fixed by the reference).
#define TB    32        // branching factor at every level == wave32 width
#define NL0   32        // level-0 nodes
#define NL1   1024      // level-1 nodes
#define NC    33824     // total channels (32 + 1024 + 32768)
#define NHW   169       // 13*13
#define NN    8         // batch
#define THRESH 0.5f

// ---------------------------------------------------------------------------
// Kernel 1: zero-fill d_out (183 MB) using CDNA5 async LDS->global streaming
// stores. Each thread owns a 16B zero slot in LDS and fires
// global_store_async_from_lds_b128 per float4 of output. ASYNCcnt tracks the
// transfers; hardware stalls issue at counter saturation, and we drain with
// s_wait_asynccnt 0 before ending.
// ---------------------------------------------------------------------------
__global__ __launch_bounds__(256) void zero_fill_async_kernel(
    float* __restrict__ out, unsigned n4, unsigned tail) {
  __shared__ __align__(16) float lzero[256 * 4];  // 16B per thread
  const unsigned tid = threadIdx.x;
  lzero[tid * 4 + 0] = 0.0f;
  lzero[tid * 4 + 1] = 0.0f;
  lzero[tid * 4 + 2] = 0.0f;
  lzero[tid * 4 + 3] = 0.0f;
  __syncthreads();  // DS writes complete (dscnt waited) before async reads

  // Generic shared pointer: low 32 bits are the wave-relative LDS byte offset
  // (SHARED aperture = {base_hi32 , lds_offset}), which is exactly what the
  // async-store VSRC operand expects.
  const unsigned lds_off = (unsigned)(uintptr_t)&lzero[tid * 4];

  const unsigned stride = gridDim.x * blockDim.x;
  for (unsigned i = blockIdx.x * blockDim.x + tid; i < n4; i += stride) {
    const unsigned voff = i * 16u;  // byte offset from SADDR base (GVS mode)
    asm volatile("global_store_async_from_lds_b128 %0, %1, %2"
                 :: "v"(voff), "v"(lds_off), "s"(out)
                 : "memory");
  }
  { // tail (unused for this problem size, kept for safety)
    const unsigned gid = blockIdx.x * blockDim.x + tid;
    if (gid < tail) out[(size_t)n4 * 4 + gid] = 0.0f;
  }
  asm volatile("s_wait_asynccnt 0x0" ::: "memory");
}

// ---------------------------------------------------------------------------
// Wave-level stats of one 32-sibling group: max-softmax prob = 1/sum(exp(x-max))
// and argmax (first occurrence, matching jnp.argmax tie-break).
// ---------------------------------------------------------------------------
__device__ __forceinline__ void group_stats(float v, int lane,
                                            float& pmax, int& amax) {
  float gmax = v;
#pragma unroll
  for (int off = 16; off > 0; off >>= 1)
    gmax = fmaxf(gmax, __shfl_xor(gmax, off, 32));

#if __has_builtin(__builtin_amdgcn_ballot_w32)
  const unsigned mm = __builtin_amdgcn_ballot_w32(v == gmax);
  amax = __ffs(mm) - 1;
#else
  int cand = (v == gmax) ? lane : 32;
#pragma unroll
  for (int off = 16; off > 0; off >>= 1)
    cand = min(cand, __shfl_xor(cand, off, 32));
  amax = cand;
#endif

  float e = __expf(v - gmax);
#pragma unroll
  for (int off = 16; off > 0; off >>= 1)
    e += __shfl_xor(e, off, 32);
  pmax = 1.0f / e;  // = max(softmax) over the group
}

// ---------------------------------------------------------------------------
// Kernel 2: one wave32 per pixel; lane j == sibling j of the current group.
// Greedy descent through 3 levels (each level's addresses depend on the
// previous argmax, so the wave walks sequentially), then lanes 0/1/2 scatter
// the three gated path probabilities.
// ---------------------------------------------------------------------------
__global__ __launch_bounds__(256) void tree_predict_kernel(
    const float* __restrict__ x, float* __restrict__ out) {
  const int wid  = (int)((blockIdx.x * blockDim.x + threadIdx.x) >> 5);
  const int lane = (int)(threadIdx.x & 31);
  if (wid >= NN * NHW) return;

  const int n = wid / NHW;
  const int s = wid - n * NHW;
  const size_t base = (size_t)n * NC * NHW + (size_t)s;
  const float* xb = x + base;

  // Level 0: channels [0, 32)
  float m0; int a0;
  group_stats(xb[(size_t)lane * NHW], lane, m0, a0);
  const bool ok0 = m0 > THRESH;

  // Level 1: group a0 -> channels [32 + 32*a0, +32)
  float p1; int a1;
  group_stats(xb[(size_t)(NL0 + a0 * TB + lane) * NHW], lane, p1, a1);
  const float m1 = m0 * p1;
  const bool ok1 = ok0 && (m1 > THRESH);
  const int idx1 = a0 * TB + a1;

  // Level 2: group idx1 -> channels [32+1024 + 32*idx1, +32)
  float p2; int a2;
  group_stats(xb[(size_t)(NL0 + NL1 + idx1 * TB + lane) * NHW], lane, p2, a2);
  const float m2 = m1 * p2;
  const bool ok2 = ok1 && (m2 > THRESH);
  const int idx2 = idx1 * TB + a2;

  float* ob = out + base;
  if (lane == 0)      ob[(size_t)a0 * NHW]                     = ok0 ? m0 : 0.0f;
  else if (lane == 1) ob[(size_t)(NL0 + idx1) * NHW]           = ok1 ? m1 : 0.0f;
  else if (lane == 2) ob[(size_t)(NL0 + NL1 + idx2) * NHW]     = ok2 ? m2 : 0.0f;
}

// ---------------------------------------------------------------------------
extern "C" void kernel_launch(void* const* d_in, const int* in_sizes, int n_in,
                              void* d_out, int out_size, void* d_ws, size_t ws_size,
                              hipStream_t stream) {
  (void)in_sizes; (void)n_in; (void)d_ws; (void)ws_size;
  const float* x = (const float*)d_in[0];
  float* out = (float*)d_out;

  const unsigned n4   = (unsigned)(out_size / 4);
  const unsigned tail = (unsigned)(out_size - (size_t)n4 * 4);

  // 183 MB zero-fill: store-bandwidth bound (~7.9 us floor at 23.3 TB/s).
  zero_fill_async_kernel<<<2048, 256, 0, stream>>>(out, n4, tail);

  // 1352 pixels -> 1352 waves -> 169 blocks of 8 waves.
  const int waves  = NN * NHW;
  const int blocks = (waves * 32 + 255) / 256;
  tree_predict_kernel<<<blocks, 256, 0, stream>>>(x, out);
}